// GCN_12575664243073
// MI455X (gfx1250) — compile-verified
//
#include <hip/hip_runtime.h>
#include <math.h>

// ---------------------------------------------------------------------------
// GCN on MI455X (gfx1250, wave32). fp32 end-to-end; GEMMs use
// V_WMMA_F32_16X16X4_F32 (memory-bound problem -> fp32 WMMA is free).
// Workspace requirement: 3*N*176*4 + 2*N*4 + (64*160+64)*4  ~= 212 MB.
// ---------------------------------------------------------------------------

#define Nn 100000
#define Ee 450000
#define Bb 64
#define SLOPE 0.01f

typedef float v2f __attribute__((ext_vector_type(2)));
typedef float v8f __attribute__((ext_vector_type(8)));

__device__ __forceinline__ float leaky(float v) { return v > 0.f ? v : SLOPE * v; }

// ---------------------------------------------------------------------------
// WMMA GEMM: C[N x M] = rowscale(A[N x K]) @ W[K x M], optional fused epilogue
// One wave handles a 16-row tile and all M/16 column tiles (MT compile-time).
// A-fragment layout (32-bit A 16x4): lane m=lane&15 is row; lanes>=16 take K+2.
// C layout: VGPR r -> row r (lanes 0-15) / row r+8 (lanes 16-31), N = lane&15.
// ---------------------------------------------------------------------------
template <int MT, bool SCALE_A, bool EPI>
__global__ __launch_bounds__(256) void gemm_wmma(
    const float* __restrict__ A, const float* __restrict__ W,
    const float* __restrict__ bias, const float* __restrict__ onorm,
    const float* __restrict__ inorm, float* __restrict__ C, int K) {
  constexpr int M = MT * 16;
  int wave = (blockIdx.x * blockDim.x + threadIdx.x) >> 5;
  if (wave >= Nn / 16) return;
  int lane = threadIdx.x & 31;
  int m = lane & 15;
  int hi = lane >> 4;  // lanes 16-31 handle K offsets +2,+3

  int row0 = wave * 16;
  const float* arow = A + (size_t)(row0 + m) * K + 2 * hi;
  float ascale = SCALE_A ? onorm[row0 + m] : 1.0f;

  v8f acc[MT];
  v8f vz = {0.f, 0.f, 0.f, 0.f, 0.f, 0.f, 0.f, 0.f};
#pragma unroll
  for (int t = 0; t < MT; ++t) acc[t] = vz;

  for (int k = 0; k < K; k += 4) {
    v2f a = *(const v2f*)(arow + k);  // A[row][k+2hi], A[row][k+2hi+1]
    a[0] *= ascale;
    a[1] *= ascale;
    const float* wk = W + (size_t)(k + 2 * hi) * M + m;
#pragma unroll
    for (int t = 0; t < MT; ++t) {
      v2f b;
      b[0] = wk[t * 16];      // W[k+2hi][t*16+m]
      b[1] = wk[t * 16 + M];  // W[k+2hi+1][t*16+m]
      acc[t] = __builtin_amdgcn_wmma_f32_16x16x4_f32(
          false, a, false, b, (short)0, acc[t], false, false);
    }
  }

  int rbase = row0 + hi * 8;
#pragma unroll
  for (int r = 0; r < 8; ++r) {
    int row = rbase + r;
    float sc = EPI ? inorm[row] : 1.0f;
    float* crow = C + (size_t)row * M + m;
#pragma unroll
    for (int t = 0; t < MT; ++t) {
      float v = acc[t][r];
      if (EPI) v = leaky(v * sc + bias[t * 16 + m]);
      crow[t * 16] = v;
    }
  }
}

// ---------------------------------------------------------------------------
// Edge scatter: agg[dst[e]] += x[src[e]] (* out_norm[src[e]] if SCALE).
// One thread per (edge, 4-column chunk); float4 gather + f32 global atomics.
// ---------------------------------------------------------------------------
template <bool SCALE>
__global__ void edge_scatter(const float* __restrict__ x, const int* __restrict__ src,
                             const int* __restrict__ dst, const float* __restrict__ onorm,
                             float* __restrict__ agg, int d) {
  int chunks = d >> 2;
  int tid = blockIdx.x * blockDim.x + threadIdx.x;
  if (tid >= Ee * chunks) return;
  int e = tid / chunks;
  int c = (tid - e * chunks) << 2;
  int s = src[e], t = dst[e];
  float4 v = *(const float4*)(x + (size_t)s * d + c);
  float sc = SCALE ? onorm[s] : 1.0f;
  float* ap = agg + (size_t)t * d + c;
  atomicAdd(ap + 0, v.x * sc);
  atomicAdd(ap + 1, v.y * sc);
  atomicAdd(ap + 2, v.z * sc);
  atomicAdd(ap + 3, v.w * sc);
}

__global__ void zero_kernel(float* __restrict__ p, int n) {
  int i = blockIdx.x * blockDim.x + threadIdx.x;
  if (i < n) p[i] = 0.f;
}

__global__ void edge_degrees(const int* __restrict__ src, const int* __restrict__ dst,
                             float* __restrict__ odeg, float* __restrict__ ideg) {
  int e = blockIdx.x * blockDim.x + threadIdx.x;
  if (e >= Ee) return;
  atomicAdd(&odeg[src[e]], 1.0f);
  atomicAdd(&ideg[dst[e]], 1.0f);
}

__global__ void deg_to_norm(float* __restrict__ o, float* __restrict__ i) {
  int n = blockIdx.x * blockDim.x + threadIdx.x;
  if (n >= Nn) return;
  o[n] = 1.0f / sqrtf(fmaxf(o[n], 1.0f));
  i[n] = 1.0f / sqrtf(fmaxf(i[n], 1.0f));
}

// epilogue for GEMM-first layers: h = leaky(agg * in_norm[row] + b[col])
__global__ void activate_epilogue(float* __restrict__ h, const float* __restrict__ inorm,
                                  const float* __restrict__ bias, int d, int total) {
  int i = blockIdx.x * blockDim.x + threadIdx.x;
  if (i >= total) return;
  int row = i / d;
  int col = i - row * d;
  h[i] = leaky(h[i] * inorm[row] + bias[col]);
}

// ---------------------------------------------------------------------------
// Graph mean-pool: LDS-staged partial sums per block, then atomics to global.
// ---------------------------------------------------------------------------
__global__ __launch_bounds__(256) void pool_sum(const float* __restrict__ h,
                                                const int* __restrict__ gid,
                                                float* __restrict__ hg,
                                                float* __restrict__ cnt) {
  __shared__ float lsum[Bb * 160];
  __shared__ float lcnt[Bb];
  for (int i = threadIdx.x; i < Bb * 160; i += blockDim.x) lsum[i] = 0.f;
  if (threadIdx.x < Bb) lcnt[threadIdx.x] = 0.f;
  __syncthreads();
  int per = (Nn + gridDim.x - 1) / gridDim.x;
  int n0 = blockIdx.x * per;
  int n1 = n0 + per;
  if (n1 > Nn) n1 = Nn;
  for (int n = n0; n < n1; ++n) {
    int g = gid[n];
    if (threadIdx.x < 160) lsum[g * 160 + threadIdx.x] += h[(size_t)n * 160 + threadIdx.x];
    if (threadIdx.x == 0) lcnt[g] += 1.f;
  }
  __syncthreads();
  for (int i = threadIdx.x; i < Bb * 160; i += blockDim.x) atomicAdd(&hg[i], lsum[i]);
  if (threadIdx.x < Bb) atomicAdd(&cnt[threadIdx.x], lcnt[threadIdx.x]);
}

// ---------------------------------------------------------------------------
// JAX threefry2x32 dropout (key = (0, seed); counter = split iota; fp32 bits).
// ---------------------------------------------------------------------------
__device__ __forceinline__ unsigned rotl32(unsigned x, unsigned r) {
  return (x << r) | (x >> (32u - r));
}

__device__ inline void threefry2x32(unsigned k0, unsigned k1, unsigned x0, unsigned x1,
                                    unsigned& o0, unsigned& o1) {
  unsigned ks0 = k0, ks1 = k1, ks2 = 0x1BD11BDAu ^ k0 ^ k1;
  x0 += ks0;
  x1 += ks1;
#define TF_R4(r0, r1, r2, r3)        \
  x0 += x1; x1 = rotl32(x1, r0); x1 ^= x0; \
  x0 += x1; x1 = rotl32(x1, r1); x1 ^= x0; \
  x0 += x1; x1 = rotl32(x1, r2); x1 ^= x0; \
  x0 += x1; x1 = rotl32(x1, r3); x1 ^= x0;
  TF_R4(13u, 15u, 26u, 6u)  x0 += ks1; x1 += ks2 + 1u;
  TF_R4(17u, 29u, 16u, 24u) x0 += ks2; x1 += ks0 + 2u;
  TF_R4(13u, 15u, 26u, 6u)  x0 += ks0; x1 += ks1 + 3u;
  TF_R4(17u, 29u, 16u, 24u) x0 += ks1; x1 += ks2 + 4u;
  TF_R4(13u, 15u, 26u, 6u)  x0 += ks2; x1 += ks0 + 5u;
#undef TF_R4
  o0 = x0;
  o1 = x1;
}

__device__ inline float jax_uniform(unsigned seed, int idx, int n) {
  int half = n >> 1;
  bool second = idx >= half;
  int i = second ? idx - half : idx;
  unsigned o0, o1;
  threefry2x32(0u, seed, (unsigned)i, (unsigned)(half + i), o0, o1);
  unsigned bits = second ? o1 : o0;
  return __uint_as_float((bits >> 9) | 0x3f800000u) - 1.0f;
}

__device__ inline float jax_dropout(float v, float keep_p, unsigned seed, int idx, int n) {
  return (jax_uniform(seed, idx, n) < keep_p) ? v / keep_p : 0.0f;
}

// ---------------------------------------------------------------------------
// MLP head: one block per graph. leaky(mean) -> 160x140 -> drop/leaky ->
// 140x10 -> drop -> sigmoid.
// ---------------------------------------------------------------------------
__global__ __launch_bounds__(192) void mlp_head(const float* __restrict__ hg,
                                                const float* __restrict__ cnt,
                                                const float* __restrict__ w1,
                                                const float* __restrict__ b1,
                                                const float* __restrict__ w2,
                                                const float* __restrict__ b2,
                                                float* __restrict__ out) {
  __shared__ float sh[160];
  __shared__ float sz[140];
  int b = blockIdx.x;
  int t = threadIdx.x;
  if (t < 160) {
    float c = fmaxf(cnt[b], 1.0f);
    sh[t] = leaky(hg[b * 160 + t] / c);
  }
  __syncthreads();
  if (t < 140) {
    float acc = b1[t];
    for (int k = 0; k < 160; ++k) acc += sh[k] * w1[k * 140 + t];
    acc = jax_dropout(acc, 0.7f, 101u, b * 140 + t, Bb * 140);
    sz[t] = leaky(acc);
  }
  __syncthreads();
  if (t < 10) {
    float acc = b2[t];
    for (int k = 0; k < 140; ++k) acc += sz[k] * w2[k * 10 + t];
    acc = jax_dropout(acc, 0.8f, 102u, b * 10 + t, Bb * 10);
    out[b * 10 + t] = 1.0f / (1.0f + expf(-acc));
  }
}

// ---------------------------------------------------------------------------
// Host orchestration
// ---------------------------------------------------------------------------
struct GemmArgs {
  const float* A;
  const float* W;
  const float* bias;
  const float* onorm;
  const float* inorm;
  float* C;
  int K;
};

template <int MT, bool SA, bool EPI>
static inline void run_gemm(const GemmArgs& g, hipStream_t s) {
  int blocks = (Nn / 16 + 7) / 8;  // 8 waves (tiles) per 256-thread block
  gemm_wmma<MT, SA, EPI><<<blocks, 256, 0, s>>>(g.A, g.W, g.bias, g.onorm, g.inorm, g.C, g.K);
}

static void gemm_dispatch(int MT, bool gemm_first, const GemmArgs& g, hipStream_t s) {
  if (gemm_first) {  // scale A rows by out_norm, no epilogue
    switch (MT) {
      case 5:  run_gemm<5, true, false>(g, s); break;
      case 6:  run_gemm<6, true, false>(g, s); break;
      case 7:  run_gemm<7, true, false>(g, s); break;
      case 8:  run_gemm<8, true, false>(g, s); break;
      case 9:  run_gemm<9, true, false>(g, s); break;
      case 10: run_gemm<10, true, false>(g, s); break;
      default: run_gemm<11, true, false>(g, s); break;
    }
  } else {  // fused in_norm + bias + leaky epilogue
    switch (MT) {
      case 5:  run_gemm<5, false, true>(g, s); break;
      case 6:  run_gemm<6, false, true>(g, s); break;
      case 7:  run_gemm<7, false, true>(g, s); break;
      case 8:  run_gemm<8, false, true>(g, s); break;
      case 9:  run_gemm<9, false, true>(g, s); break;
      case 10: run_gemm<10, false, true>(g, s); break;
      default: run_gemm<11, false, true>(g, s); break;
    }
  }
}

extern "C" void kernel_launch(void* const* d_in, const int* in_sizes, int n_in,
                              void* d_out, int out_size, void* d_ws, size_t ws_size,
                              hipStream_t stream) {
  (void)in_sizes; (void)n_in; (void)out_size; (void)ws_size;
  const float* inputs = (const float*)d_in[0];
  const int* src = (const int*)d_in[2];
  const int* dst = (const int*)d_in[3];
  const int* gid = (const int*)d_in[4];
  const float* cw[11];
  const float* cb[11];
  for (int i = 0; i < 11; ++i) {
    cw[i] = (const float*)d_in[5 + i];
    cb[i] = (const float*)d_in[16 + i];
  }
  const float* w1 = (const float*)d_in[27];
  const float* b1 = (const float*)d_in[28];
  const float* w2 = (const float*)d_in[29];
  const float* b2 = (const float*)d_in[30];
  float* out = (float*)d_out;

  // workspace carve-up
  const size_t S = (size_t)Nn * 176;  // floats per ping-pong buffer
  float* p = (float*)d_ws;
  float* bufs[3] = {p, p + S, p + 2 * S};
  float* onorm = p + 3 * S;
  float* inorm = onorm + Nn;
  float* hg = inorm + Nn;        // Bb*160
  float* cnt = hg + Bb * 160;    // Bb

  const int dims[12] = {128, 80, 160, 112, 160, 176, 96, 144, 96, 128, 96, 160};
  const bool gemm_first[11] = {true, false, true, false, false, true, false, true, false, true, false};

  // degrees -> norms
  zero_kernel<<<(2 * Nn + 255) / 256, 256, 0, stream>>>(onorm, 2 * Nn);  // onorm+inorm contiguous
  edge_degrees<<<(Ee + 255) / 256, 256, 0, stream>>>(src, dst, onorm, inorm);
  deg_to_norm<<<(Nn + 255) / 256, 256, 0, stream>>>(onorm, inorm);

  const float* h = inputs;
  for (int l = 0; l < 11; ++l) {
    int din = dims[l], dout = dims[l + 1];
    float* f1 = nullptr;
    float* f2 = nullptr;
    for (int i = 0; i < 3; ++i) {
      if ((const float*)bufs[i] == h) continue;
      if (!f1) f1 = bufs[i];
      else if (!f2) f2 = bufs[i];
    }
    if (gemm_first[l]) {
      // y = (h * onorm) @ W  -> scatter y -> epilogue
      GemmArgs g{h, cw[l], nullptr, onorm, nullptr, f1, din};
      gemm_dispatch(dout / 16, true, g, stream);
      zero_kernel<<<(Nn * dout + 255) / 256, 256, 0, stream>>>(f2, Nn * dout);
      int threads = Ee * (dout / 4);
      edge_scatter<false><<<(threads + 255) / 256, 256, 0, stream>>>(f1, src, dst, nullptr, f2, dout);
      activate_epilogue<<<(Nn * dout + 255) / 256, 256, 0, stream>>>(f2, inorm, cb[l], dout, Nn * dout);
      h = f2;
    } else {
      // scatter (h * onorm) -> GEMM with fused in_norm+bias+leaky
      zero_kernel<<<(Nn * din + 255) / 256, 256, 0, stream>>>(f1, Nn * din);
      int threads = Ee * (din / 4);
      edge_scatter<true><<<(threads + 255) / 256, 256, 0, stream>>>(h, src, dst, onorm, f1, din);
      GemmArgs g{f1, cw[l], cb[l], nullptr, inorm, f2, din};
      gemm_dispatch(dout / 16, false, g, stream);
      h = f2;
    }
  }

  // mean-pool over graphs, then MLP head
  zero_kernel<<<(Bb * 160 + Bb + 255) / 256, 256, 0, stream>>>(hg, Bb * 160 + Bb);  // hg+cnt contiguous
  pool_sum<<<120, 256, 0, stream>>>(h, gid, hg, cnt);
  mlp_head<<<Bb, 192, 0, stream>>>(hg, cnt, w1, b1, w2, b2, out);
}